// GNNEncoder_50036368998569
// MI455X (gfx1250) — compile-verified
//
#include <hip/hip_runtime.h>

// GNN encoder for MI455X (gfx1250, wave32).
// GEMMs use V_WMMA_F32_16X16X4_F32 (fp32 matrix path, exact reference math).
// Edge aggregation is the HBM-bound part (~2.6 GB total -> ~110us at 23.3 TB/s);
// it uses coalesced float4 gathers of h[src] + global_atomic_add_f32 scatters.

typedef __attribute__((ext_vector_type(2))) float v2f;
typedef __attribute__((ext_vector_type(8))) float v8f;

#define BN_EPS 1e-5f

// ---------------- degree / normalization precompute ----------------

__global__ void edge_deg_kernel(const long long* __restrict__ ei,
                                const float* __restrict__ ew,
                                float* __restrict__ deg, int E) {
  int e = blockIdx.x * blockDim.x + threadIdx.x;
  if (e < E) atomicAdd(&deg[ei[(size_t)E + e]], ew[e]);
}

__global__ void dis_kernel(float* __restrict__ deg, int N) {
  int i = blockIdx.x * blockDim.x + threadIdx.x;
  // self-loop contributes weight 1, so deg+1 > 0 always
  if (i < N) deg[i] = rsqrtf(deg[i] + 1.0f);
}

__global__ void norm_kernel(const long long* __restrict__ ei,
                            const float* __restrict__ ew,
                            const float* __restrict__ dis,
                            float* __restrict__ nrm, int E) {
  int e = blockIdx.x * blockDim.x + threadIdx.x;
  if (e < E) nrm[e] = dis[ei[e]] * ew[e] * dis[ei[(size_t)E + e]];
}

// ---------------- fp32 GEMM via V_WMMA_F32_16X16X4_F32 ----------------
// Block = 256 threads = 8 waves. W (K x C, row-major) staged in LDS.
// Each wave computes one 16x16 tile of Y = X @ W.

__global__ void gemm_wmma_f32(const float* __restrict__ X,
                              const float* __restrict__ Wg,
                              float* __restrict__ Y,
                              int N, int K, int C, int nTiles) {
  extern __shared__ float Wlds[];
  const int tid = threadIdx.x;
  for (int i = tid; i < K * C; i += blockDim.x) Wlds[i] = Wg[i];
  __syncthreads();

  const int wave = tid >> 5;
  const int lane = tid & 31;
  const int tile = blockIdx.x * 8 + wave;
  if (tile >= nTiles) return;            // wave-uniform exit (EXEC stays full)

  const int nCT  = C >> 4;
  const int row0 = (tile / nCT) << 4;
  const int col0 = (tile % nCT) << 4;

  // A layout (16x4 f32): lanes 0-15 -> M=lane, K=k+0/k+1; lanes 16-31 -> K=k+2/k+3
  const int m  = lane & 15;
  const int kh = (lane >> 4) << 1;
  int row = row0 + m; if (row >= N) row = N - 1;   // N is a multiple of 16 here; safety clamp

  const float* xp = X + (size_t)row * K + kh;
  const float* wp = Wlds + (size_t)kh * C + col0 + m;

  v8f acc = {0.f, 0.f, 0.f, 0.f, 0.f, 0.f, 0.f, 0.f};
  for (int k = 0; k < K; k += 4) {
    v2f a, b;
    a.x = xp[k];
    a.y = xp[k + 1];
    b.x = wp[(size_t)k * C];
    b.y = wp[(size_t)(k + 1) * C];
    acc = __builtin_amdgcn_wmma_f32_16x16x4_f32(
        /*neg_a=*/false, a, /*neg_b=*/false, b,
        /*c_mod=*/(short)0, acc, /*reuse_a=*/false, /*reuse_b=*/false);
  }

  // C/D layout: VGPR v, lanes 0-15 -> M=v; lanes 16-31 -> M=v+8; N=lane&15
  const int mb = (lane >> 4) << 3;
  float* yp = Y + (size_t)(row0 + mb) * C + col0 + (lane & 15);
#pragma unroll
  for (int v = 0; v < 8; ++v) {
    if (row0 + mb + v < N) yp[(size_t)v * C] = acc[v];
  }
}

// ---------------- aggregation ----------------

// agg[i][c] = dis[i]^2 * h[i][c] + b[c]   (self-loop term + bias)
__global__ void init_agg_kernel(const float* __restrict__ h,
                                const float* __restrict__ dis,
                                const float* __restrict__ bias,
                                float* __restrict__ agg, int N, int C) {
  const int q = C >> 2;
  int t = blockIdx.x * blockDim.x + threadIdx.x;
  if (t >= N * q) return;
  const int i  = t / q;
  const int c4 = (t % q) << 2;
  const float d = dis[i];
  const float w = d * d;
  const float4 hv = *(const float4*)(h + (size_t)i * C + c4);
  const float4 bv = *(const float4*)(bias + c4);
  float4 o;
  o.x = w * hv.x + bv.x;
  o.y = w * hv.y + bv.y;
  o.z = w * hv.z + bv.z;
  o.w = w * hv.w + bv.w;
  *(float4*)(agg + (size_t)i * C + c4) = o;
}

// agg[dst][c] += norm[e] * h[src][c]  (the HBM-bound scatter)
__global__ void edge_agg_kernel(const long long* __restrict__ ei,
                                const float* __restrict__ nrm,
                                const float* __restrict__ h,
                                float* __restrict__ agg, int E, int C) {
  const int q = C >> 2;
  int t = blockIdx.x * blockDim.x + threadIdx.x;
  if (t >= E * q) return;
  const int e  = t / q;
  const int c4 = (t % q) << 2;
  const long long s = ei[e];
  const long long d = ei[(size_t)E + e];
  const float w = nrm[e];
  const float4 hv = *(const float4*)(h + (size_t)s * C + c4);
  float* o = agg + (size_t)d * C + c4;
  atomicAdd(o + 0, w * hv.x);
  atomicAdd(o + 1, w * hv.y);
  atomicAdd(o + 2, w * hv.z);
  atomicAdd(o + 3, w * hv.w);
}

// ---------------- batchnorm (+ReLU) ----------------

// blockDim.x == C; thread c accumulates sum/sumsq over a row chunk.
__global__ void bn_stats_kernel(const float* __restrict__ xin,
                                float* __restrict__ stats,
                                int N, int C, int rowsPerBlock) {
  const int c  = threadIdx.x;
  const int r0 = blockIdx.x * rowsPerBlock;
  int rend = r0 + rowsPerBlock; if (rend > N) rend = N;
  float s = 0.f, s2 = 0.f;
  for (int r = r0; r < rend; ++r) {
    float v = xin[(size_t)r * C + c];
    s += v; s2 += v * v;
  }
  atomicAdd(&stats[c], s);
  atomicAdd(&stats[C + c], s2);
}

__global__ void bn_apply_kernel(const float* __restrict__ xin,
                                const float* __restrict__ stats,
                                const float* __restrict__ g,
                                const float* __restrict__ be,
                                float* __restrict__ out, int N, int C) {
  int t = blockIdx.x * blockDim.x + threadIdx.x;
  if (t >= N * C) return;
  const int c = t % C;
  const float inv_n = 1.0f / (float)N;
  const float m   = stats[c] * inv_n;
  const float var = stats[C + c] * inv_n - m * m;
  float v = (xin[t] - m) * rsqrtf(var + BN_EPS) * g[c] + be[c];
  out[t] = v > 0.f ? v : 0.f;
}

// ---------------- orchestration ----------------

static void run_layer(const float* in, const float* W, const float* b,
                      const float* g, const float* be,
                      float* hbuf, float* aggbuf, float* outbuf,
                      const long long* ei, const float* nrm, const float* dis,
                      float* stats, int N, int Cin, int Cout, int E,
                      hipStream_t stream) {
  const int nTiles = (N / 16) * (Cout / 16);
  const int grid   = (nTiles + 7) / 8;
  const size_t smem = (size_t)Cin * Cout * sizeof(float);
  gemm_wmma_f32<<<grid, 256, smem, stream>>>(in, W, hbuf, N, Cin, Cout, nTiles);

  const int q = Cout / 4;
  init_agg_kernel<<<(N * q + 255) / 256, 256, 0, stream>>>(hbuf, dis, b, aggbuf, N, Cout);
  edge_agg_kernel<<<(E * q + 255) / 256, 256, 0, stream>>>(ei, nrm, hbuf, aggbuf, E, Cout);

  hipMemsetAsync(stats, 0, (size_t)2 * Cout * sizeof(float), stream);
  bn_stats_kernel<<<(N + 127) / 128, Cout, 0, stream>>>(aggbuf, stats, N, Cout, 128);
  bn_apply_kernel<<<(N * Cout + 255) / 256, 256, 0, stream>>>(aggbuf, stats, g, be, outbuf, N, Cout);
}

extern "C" void kernel_launch(void* const* d_in, const int* in_sizes, int n_in,
                              void* d_out, int out_size, void* d_ws, size_t ws_size,
                              hipStream_t stream) {
  const float*     x   = (const float*)d_in[0];
  const long long* ei  = (const long long*)d_in[1];   // edge_index int64, [2, E]
  const float*     ew  = (const float*)d_in[2];
  const float *W0 = (const float*)d_in[3],  *b0 = (const float*)d_in[4];
  const float *g0 = (const float*)d_in[5],  *be0 = (const float*)d_in[6];
  const float *W1 = (const float*)d_in[7],  *b1 = (const float*)d_in[8];
  const float *g1 = (const float*)d_in[9],  *be1 = (const float*)d_in[10];
  const float *Wf = (const float*)d_in[11], *bf = (const float*)d_in[12];
  const float *gf = (const float*)d_in[13], *bef = (const float*)d_in[14];

  const int E   = in_sizes[2];
  const int H   = in_sizes[4];          // 128
  const int Cin = in_sizes[3] / H;      // 128
  const int L   = in_sizes[12];         // 64
  const int N   = in_sizes[0] / Cin;    // 50000

  // workspace layout (all 16B-aligned chunks)
  float* bufA  = (float*)d_ws;                    // [N, H]   gemm output
  float* bufB  = bufA + (size_t)N * H;            // [N, H]   agg / bn / next input
  float* dis   = bufB + (size_t)N * H;            // [N]      (deg in-place -> rsqrt)
  float* nrm   = dis + N;                         // [E]
  float* stats = nrm + E;                         // [2*H]

  // degree + symmetric normalization (reused by all 3 layers)
  hipMemsetAsync(dis, 0, (size_t)N * sizeof(float), stream);
  edge_deg_kernel<<<(E + 255) / 256, 256, 0, stream>>>(ei, ew, dis, E);
  dis_kernel<<<(N + 255) / 256, 256, 0, stream>>>(dis, N);
  norm_kernel<<<(E + 255) / 256, 256, 0, stream>>>(ei, ew, dis, nrm, E);

  // layer 0: x -> bufB
  run_layer(x,    W0, b0, g0, be0, bufA, bufB, bufB, ei, nrm, dis, stats, N, Cin, H, E, stream);
  // layer 1: bufB -> bufB
  run_layer(bufB, W1, b1, g1, be1, bufA, bufB, bufB, ei, nrm, dis, stats, N, H,   H, E, stream);
  // final layer: bufB -> d_out (agg + BN in place on d_out)
  float* out = (float*)d_out;
  run_layer(bufB, Wf, bf, gf, bef, bufA, out, out, ei, nrm, dis, stats, N, H, L, E, stream);
}